// ImgfeatIbrnet_26542897889297
// MI455X (gfx1250) — compile-verified
//
#include <hip/hip_runtime.h>
#include <hip/hip_bf16.h>
#include <stdint.h>

// ---------------------------------------------------------------------------
// IBRNet feature/rgb projection for MI455X (gfx1250).
// Memory-bound gather+elementwise kernel (AI ~ 1.5 FLOP/byte): no WMMA role.
// CDNA5 paths: wave32 blocks, LDS-staged coalesced output written with
// global_store_async_from_lds_b128 (ASYNCcnt), global_prefetch_b8.
// LDS tile uses a per-point 16B-chunk rotation to cut DS bank conflicts
// from 8-way (stride 40 dwords, gcd(40,64)=8) to ~2-way worst case.
// ---------------------------------------------------------------------------

#define TPB      256
#define S_VIEWS  8
#define NPTS     (2048 * 64)   // NR*NS = 131072
#define IMG_H    512
#define IMG_W    512
#define FEAT_C   32
#define FEAT_H   128
#define FEAT_W   128
#define OUTC     40            // 3 rgb + 32 feat + 3 diff + 1 dot + 1 mask
#define NCHUNK   (OUTC / 4)    // 10 x 16B chunks per point

#define USE_ASYNC_LDS_STORE 1

// -------------------------------------------------------------------
// Pre-kernel: cam_center[s] = inv(ext_s)[:3,3] via adjugate column 3.
// inv[i][3] = (-1)^(3+i) * minor(3,i) / det ;  det expanded along row 3.
// -------------------------------------------------------------------
__global__ void cam_center_kernel(const float* __restrict__ exts,
                                  float* __restrict__ cc) {
  int s = threadIdx.x;
  if (s >= S_VIEWS) return;
  float m[16];
#pragma unroll
  for (int i = 0; i < 16; ++i) m[i] = exts[s * 16 + i];

  auto det3 = [&](int c0, int c1, int c2) -> float {
    float a = m[0*4+c0], b = m[0*4+c1], c = m[0*4+c2];
    float d = m[1*4+c0], e = m[1*4+c1], f = m[1*4+c2];
    float g = m[2*4+c0], h = m[2*4+c1], i = m[2*4+c2];
    return a*(e*i - f*h) - b*(d*i - f*g) + c*(d*h - e*g);
  };
  float M30 = det3(1,2,3);
  float M31 = det3(0,2,3);
  float M32 = det3(0,1,3);
  float M33 = det3(0,1,2);
  float det = -m[12]*M30 + m[13]*M31 - m[14]*M32 + m[15]*M33;
  float inv_det = 1.0f / det;
  cc[s*4 + 0] = -M30 * inv_det;
  cc[s*4 + 1] =  M31 * inv_det;
  cc[s*4 + 2] = -M32 * inv_det;
  cc[s*4 + 3] = 0.0f;
}

// -------------------------------------------------------------------
// Main kernel: one thread per (view, point). Blocks: (NPTS/256, S).
// -------------------------------------------------------------------
__global__ __launch_bounds__(TPB) void ibr_project_kernel(
    const float* __restrict__ xyz,       // [P,3]
    const float* __restrict__ viewdir,   // [P,3]
    const float* __restrict__ src_inps,  // [S,3,512,512]
    const float* __restrict__ src_exts,  // [S,4,4]
    const float* __restrict__ src_ixts,  // [S,3,3]
    const float* __restrict__ img_feat,  // [S,32,128,128]
    const float* __restrict__ cam_cc,    // [S,4]  (from pre-kernel)
    float* __restrict__ out)             // [S,P,40]
{
  __shared__ float tile[TPB * OUTC];     // 40 KB, chunk-rotated layout

  const int tid = threadIdx.x;
  const int s   = blockIdx.y;
  const int p   = blockIdx.x * TPB + tid;   // exact: NPTS % TPB == 0

  // Rotated LDS address for (point=tid, channel=c):
  //   chunk = c/4, rot_chunk = (chunk + tid) % 10, dword = ... + (c&3)
  const int ptm = tid % NCHUNK;   // tid % 10
  auto taddr = [&](int c) -> int {
    int r = (c >> 2) + ptm;
    r = (r >= NCHUNK) ? r - NCHUNK : r;
    return (tid * NCHUNK + r) * 4 + (c & 3);
  };

  // View-uniform constants (scalar-cached: address depends only on blockIdx.y)
  const float* E = src_exts + s * 16;
  const float* K = src_ixts + s * 9;
  const float ccx = cam_cc[s*4+0], ccy = cam_cc[s*4+1], ccz = cam_cc[s*4+2];

  // Streamed per-point data (+ prefetch next tile: emits global_prefetch_b8)
  const float x  = xyz[p*3+0],     y  = xyz[p*3+1],     z  = xyz[p*3+2];
  const float vx = viewdir[p*3+0], vy = viewdir[p*3+1], vz = viewdir[p*3+2];
  __builtin_prefetch(xyz + (size_t)(p + TPB) * 3, 0, 3);
  __builtin_prefetch(viewdir + (size_t)(p + TPB) * 3, 0, 3);

  // ---- direction features -------------------------------------------------
  float px = ccx - x, py = ccy - y, pz = ccz - z;
  float pinv = 1.0f / (sqrtf(px*px + py*py + pz*pz) + 1e-6f);
  px *= pinv; py *= pinv; pz *= pinv;
  const float tx = -vx, ty = -vy, tz = -vz;
  const float dotv = tx*px + ty*py + tz*pz;
  float dx = tx - px, dy = ty - py, dz = tz - pz;
  float dinv = 1.0f / (sqrtf(dx*dx + dy*dy + dz*dz) + 1e-6f);
  dx *= dinv; dy *= dinv; dz *= dinv;

  // ---- projection ---------------------------------------------------------
  const float c0 = E[0]*x + E[1]*y + E[2]*z  + E[3];
  const float c1 = E[4]*x + E[5]*y + E[6]*z  + E[7];
  const float c2 = E[8]*x + E[9]*y + E[10]*z + E[11];
  const float q0 = K[0]*c0 + K[1]*c1 + K[2]*c2;
  const float q1 = K[3]*c0 + K[4]*c1 + K[5]*c2;
  const float q2 = K[6]*c0 + K[7]*c1 + K[8]*c2;
  const float zc  = fmaxf(q2, 1e-6f);
  const float gx  = (q0 / zc) * (1.0f / (IMG_W - 1));
  const float gy  = (q1 / zc) * (1.0f / (IMG_H - 1));
  const float msk = (gx >= 0.0f && gx <= 1.0f && gy >= 0.0f && gy <= 1.0f &&
                     q2 >= 1e-6f) ? 1.0f : 0.0f;
  const float gridx = gx * 2.0f - 1.0f;
  const float gridy = gy * 2.0f - 1.0f;

  // ---- bilinear: RGB (512x512), folds inps = src*0.5+0.5 ------------------
  {
    float ix = fminf(fmaxf((gridx + 1.0f) * 0.5f * (IMG_W - 1), 0.0f), (float)(IMG_W - 1));
    float iy = fminf(fmaxf((gridy + 1.0f) * 0.5f * (IMG_H - 1), 0.0f), (float)(IMG_H - 1));
    float fx0 = floorf(ix), fy0 = floorf(iy);
    float wx = ix - fx0, wy = iy - fy0;
    int x0 = (int)fx0, y0 = (int)fy0;
    int x1 = min(x0 + 1, IMG_W - 1);
    int y1 = min(y0 + 1, IMG_H - 1);
    float w00 = (1.0f - wx) * (1.0f - wy), w10 = wx * (1.0f - wy);
    float w01 = (1.0f - wx) * wy,          w11 = wx * wy;
    int i00 = y0 * IMG_W + x0, i10 = y0 * IMG_W + x1;
    int i01 = y1 * IMG_W + x0, i11 = y1 * IMG_W + x1;
    const float* img = src_inps + (size_t)s * 3 * IMG_H * IMG_W;
#pragma unroll
    for (int c = 0; c < 3; ++c) {
      const float* pl = img + (size_t)c * (IMG_H * IMG_W);
      float v = pl[i00]*w00 + pl[i10]*w10 + pl[i01]*w01 + pl[i11]*w11;
      tile[taddr(c)] = v * 0.5f + 0.5f;
    }
  }

  // ---- bilinear: features (128x128, 32 ch) --------------------------------
  {
    float ix = fminf(fmaxf((gridx + 1.0f) * 0.5f * (FEAT_W - 1), 0.0f), (float)(FEAT_W - 1));
    float iy = fminf(fmaxf((gridy + 1.0f) * 0.5f * (FEAT_H - 1), 0.0f), (float)(FEAT_H - 1));
    float fx0 = floorf(ix), fy0 = floorf(iy);
    float wx = ix - fx0, wy = iy - fy0;
    int x0 = (int)fx0, y0 = (int)fy0;
    int x1 = min(x0 + 1, FEAT_W - 1);
    int y1 = min(y0 + 1, FEAT_H - 1);
    float w00 = (1.0f - wx) * (1.0f - wy), w10 = wx * (1.0f - wy);
    float w01 = (1.0f - wx) * wy,          w11 = wx * wy;
    int i00 = y0 * FEAT_W + x0, i10 = y0 * FEAT_W + x1;
    int i01 = y1 * FEAT_W + x0, i11 = y1 * FEAT_W + x1;
    const float* ft = img_feat + (size_t)s * FEAT_C * FEAT_H * FEAT_W;
    __builtin_prefetch(ft + i00, 0, 3);
    __builtin_prefetch(ft + i01, 0, 3);
#pragma unroll 8
    for (int c = 0; c < FEAT_C; ++c) {
      const float* pl = ft + (size_t)c * (FEAT_H * FEAT_W);
      float v = pl[i00]*w00 + pl[i10]*w10 + pl[i01]*w01 + pl[i11]*w11;
      tile[taddr(3 + c)] = v;
    }
  }

  tile[taddr(35)] = dx;
  tile[taddr(36)] = dy;
  tile[taddr(37)] = dz;
  tile[taddr(38)] = dotv;
  tile[taddr(39)] = msk;

  __syncthreads();

  // ---- coalesced tile writeback: LDS -> global ----------------------------
  // Block output chunk is contiguous: 256 points * 160 B = 40960 B.
  // Global 16B-chunk index g is linear; LDS source undoes the rotation:
  //   pt = g/10, j = g%10, lds_chunk = pt*10 + (j + pt) % 10.
  float* gbase = out + ((size_t)s * NPTS + (size_t)blockIdx.x * TPB) * OUTC;

#if USE_ASYNC_LDS_STORE
  {
    // LDS byte offset = low 32 bits of the generic __shared__ address
    // (flat aperture encoding).
    uint32_t lds_base = (uint32_t)(uintptr_t)(&tile[0]);
    uint64_t saddr    = (uint64_t)(uintptr_t)gbase;
#pragma unroll
    for (int k = 0; k < NCHUNK; ++k) {
      int g   = tid + k * TPB;
      int pt  = g / NCHUNK;
      int j   = g - pt * NCHUNK;
      int r   = j + (pt % NCHUNK);
      r = (r >= NCHUNK) ? r - NCHUNK : r;
      uint32_t goff  = (uint32_t)g * 16u;                        // global byte off
      uint32_t laddr = lds_base + (uint32_t)(pt * NCHUNK + r) * 16u;
      asm volatile("global_store_async_from_lds_b128 %0, %1, %2"
                   :: "v"(goff), "v"(laddr), "s"(saddr)
                   : "memory");
    }
    asm volatile("s_wait_asynccnt 0" ::: "memory");
  }
#else
  {
    const float4* lsrc = (const float4*)tile;
    float4*       gdst = (float4*)gbase;
#pragma unroll
    for (int k = 0; k < NCHUNK; ++k) {
      int g  = tid + k * TPB;
      int pt = g / NCHUNK;
      int j  = g - pt * NCHUNK;
      int r  = j + (pt % NCHUNK);
      r = (r >= NCHUNK) ? r - NCHUNK : r;
      gdst[g] = lsrc[pt * NCHUNK + r];
    }
  }
#endif
}

extern "C" void kernel_launch(void* const* d_in, const int* in_sizes, int n_in,
                              void* d_out, int out_size, void* d_ws, size_t ws_size,
                              hipStream_t stream) {
  const float* xyz      = (const float*)d_in[0];
  const float* viewdir  = (const float*)d_in[1];
  const float* src_inps = (const float*)d_in[2];
  const float* src_exts = (const float*)d_in[3];
  const float* src_ixts = (const float*)d_in[4];
  const float* img_feat = (const float*)d_in[5];
  float*       out      = (float*)d_out;
  float*       cam_cc   = (float*)d_ws;   // 8 * 4 floats

  hipLaunchKernelGGL(cam_center_kernel, dim3(1), dim3(32), 0, stream,
                     src_exts, cam_cc);

  dim3 grid(NPTS / TPB, S_VIEWS);
  hipLaunchKernelGGL(ibr_project_kernel, grid, dim3(TPB), 0, stream,
                     xyz, viewdir, src_inps, src_exts, src_ixts, img_feat,
                     cam_cc, out);
}